// DistiledRegionLoss_6244882448539
// MI455X (gfx1250) — compile-verified
//
#include <hip/hip_runtime.h>
#include <math.h>

// ---- problem constants (from reference) ----
#define NBATCH   64
#define NHH      128
#define NWW      128
#define HWSZ     (NHH * NWW)          // 16384
#define NCHS     20                    // NA*(2K+1+NC) == NA*(19+NC)
#define KP       9
#define NTGT     50
#define NLANE    21                    // NL = 2K+3
#define OBJ_W    5.0f
#define NOOBJ_W  1.0f
#define SIL_TH   0.6f
#define PRETRAIN 15
#define IMW      640.0f
#define IMH      480.0f
#define DTHR     80.0f
#define SHARPC   2.0f
#define INV_EM1  0.15651764274966565f  // 1/(e^2 - 1)

typedef __attribute__((ext_vector_type(2))) float v2f;
typedef __attribute__((ext_vector_type(8))) float v8f;

// fast sigmoid: v_exp_f32 path (1-2 ulp, fine for a loss scalar)
__device__ __forceinline__ float sigm(float x) { return 1.0f / (1.0f + __expf(-x)); }

// ---------------------------------------------------------------------------
// Kernel Z: zero the coord-flag array (harness poisons ws with 0xAA)
// ---------------------------------------------------------------------------
__global__ void zero_kernel(unsigned int* __restrict__ p, int nWords) {
    int i = blockIdx.x * 256 + threadIdx.x;
    if (i < nWords) p[i] = 0u;
}

// ---------------------------------------------------------------------------
// Kernel A: parse targets. One thread per batch image.
//   valid = cumprod(tgt[:,:,1] != 0); scatter coord flags at (gj,gi);
//   for batch nB-1, emit keypoint table (64 padded targets) + valid flags.
// ---------------------------------------------------------------------------
__global__ void target_kernel(const float* __restrict__ tgt,
                              unsigned char* __restrict__ flagArr,
                              float2* __restrict__ gt,
                              float* __restrict__ validf) {
    int b = threadIdx.x;
    if (b >= NBATCH) return;
    const float* tb = tgt + b * NTGT * NLANE;
    bool v = true;
    bool last = (b == NBATCH - 1);
    for (int t = 0; t < NTGT; ++t) {
        float c1 = tb[t * NLANE + 1];
        v = v && (c1 != 0.0f);
        if (last) {
            validf[t] = v ? 1.0f : 0.0f;
            for (int k = 0; k < KP; ++k) {
                float2 g;
                g.x = tb[t * NLANE + 1 + 2 * k];
                g.y = tb[t * NLANE + 2 + 2 * k];
                gt[t * KP + k] = g;
            }
        }
        if (v) {
            float c2 = tb[t * NLANE + 2];
            int gi = (int)floorf(c1 * (float)NWW);
            int gj = (int)floorf(c2 * (float)NHH);
            if (gi >= 0 && gi < NWW && gj >= 0 && gj < NHH)
                flagArr[b * HWSZ + gj * NWW + gi] = 1;   // duplicate writes of 1 are fine
        }
    }
    if (last) {
        for (int t = NTGT; t < 64; ++t) {
            validf[t] = 0.0f;
            for (int k = 0; k < KP; ++k) { float2 z; z.x = 0.f; z.y = 0.f; gt[t * KP + k] = z; }
        }
    }
}

// ---------------------------------------------------------------------------
// Kernel B: cur_confs for batch nB-1 via V_WMMA_F32_16X16X4_F32.
//   dist^2(cell,target) = (X^2+Y^2) + [X Y 1] . [-2Gx, -2Gy, Gx^2+Gy^2]
//   -> D = A(16x4) x B(4x16) + C(row-broadcast X^2+Y^2).
// One wave per 16-cell group (same image row).
// Loop nest: k OUTER (rolled) so exactly one A/C fragment is live (no spills),
// tile INNER (unrolled x4) accumulating into 4 resident v8f fragments.
// 72 ds_bpermutes total per wave; epilogue is branchless v_sqrt/v_exp.
// ---------------------------------------------------------------------------
__global__ void curconf_kernel(const float* __restrict__ out,
                               const float2* __restrict__ gt,
                               const float* __restrict__ validf,
                               float* __restrict__ curconf) {
    const int lane  = threadIdx.x & 31;
    const int wave  = threadIdx.x >> 5;
    const int group = blockIdx.x * 8 + wave;     // 0..1023
    const int h  = group >> 3;
    const int w0 = (group & 7) << 4;
    const int n  = lane & 15;                    // row for A / column for B,C
    const int hi = lane >> 4;                    // 0: K=0/1 + C rows 0-7, 1: K=2/3 + rows 8-15
    const int w  = w0 + n;

    const float* base = out + (size_t)(NBATCH - 1) * NCHS * HWSZ + h * NWW + w;
    const float sx = IMW / (float)NWW;           // 5.0
    const float sy = IMH / (float)NHH;           // 3.75

    // per-lane valid/mean factors for its 4 target columns
    float vf[4];
#pragma unroll
    for (int tile = 0; tile < 4; ++tile)
        vf[tile] = validf[tile * 16 + n] * (1.0f / (float)KP);

    v8f acc[4];
#pragma unroll
    for (int tile = 0; tile < 4; ++tile) acc[tile] = (v8f){};

    for (int k = 0; k < KP; ++k) {               // rolled: one av/c0 live at a time
        float xv = base[(2 * k) * HWSZ];
        float yv = base[(2 * k + 1) * HWSZ];
        if (k == 0) { xv = sigm(xv); yv = sigm(yv); }
        float Xf = (xv + (float)w) * sx;
        float Yf = (yv + (float)h) * sy;
        float Rf = Xf * Xf + Yf * Yf;
        v2f av;
        av.x = hi ? 1.0f : Xf;                   // A: K0 -> X, K2 -> 1
        av.y = hi ? 0.0f : Yf;                   // A: K1 -> Y, K3 -> 0
        v8f c0;
#pragma unroll
        for (int j = 0; j < 8; ++j)
            c0[j] = __shfl(Rf, (hi << 3) + j, 32);   // C rows per layout halves

#pragma unroll
        for (int tile = 0; tile < 4; ++tile) {
            float2 g = gt[(tile * 16 + n) * KP + k];
            float Gx = g.x * IMW;
            float Gy = g.y * IMH;
            v2f bv;
            bv.x = hi ? (Gx * Gx + Gy * Gy) : (-2.0f * Gx);  // B: K0 / K2
            bv.y = hi ? 0.0f : (-2.0f * Gy);                  // B: K1 / K3
            v8f d = __builtin_amdgcn_wmma_f32_16x16x4_f32(
                false, av, false, bv, (short)0, c0, false, false);
#pragma unroll
            for (int j = 0; j < 8; ++j) {
                float dist = __builtin_amdgcn_sqrtf(fmaxf(d[j], 0.0f));  // v_sqrt_f32
                float e = __expf(SHARPC - dist * (SHARPC / DTHR));       // v_exp_f32
                float s = (e - 1.0f) * INV_EM1;
                acc[tile][j] += (dist < DTHR) ? s : 0.0f;                // cndmask
            }
        }
    }

    float maxacc[8];
#pragma unroll
    for (int j = 0; j < 8; ++j) maxacc[j] = 0.0f;
#pragma unroll
    for (int tile = 0; tile < 4; ++tile)
#pragma unroll
        for (int j = 0; j < 8; ++j)
            maxacc[j] = fmaxf(maxacc[j], acc[tile][j] * vf[tile]);

    // max over the 16 target-columns (lanes within each half)
#pragma unroll
    for (int m = 1; m < 16; m <<= 1) {
#pragma unroll
        for (int j = 0; j < 8; ++j)
            maxacc[j] = fmaxf(maxacc[j], __shfl_xor(maxacc[j], m, 32));
    }
    if (n == 0) {                                 // lanes 0 (rows 0-7) and 16 (rows 8-15)
#pragma unroll
        for (int j = 0; j < 8; ++j)
            curconf[group * 16 + (hi << 3) + j] = maxacc[j];
    }
}

// ---------------------------------------------------------------------------
// Kernel C: per-cell conf loss + (rare) coord-masked xy loss, block partials.
// Only channel 18 of both tensors is streamed (~8.4 MB -> bandwidth bound);
// x/y channels touched only at scatter cells (<=50 per image).
// ---------------------------------------------------------------------------
__global__ void loss_kernel(const float* __restrict__ out,
                            const float* __restrict__ dt,
                            const unsigned char* __restrict__ flagArr,
                            const float* __restrict__ curconf,
                            const int* __restrict__ epochp,
                            float* __restrict__ partials) {
    __shared__ float sdata[256];
    const int idx = blockIdx.x * 256 + threadIdx.x;     // 0 .. nB*HW-1
    const int b  = idx >> 14;
    const int hw = idx & (HWSZ - 1);
    const float* ob = out + (size_t)b * NCHS * HWSZ;
    const float* db = dt  + (size_t)b * NCHS * HWSZ;

    const unsigned char fl = flagArr[idx];
    float cm;
    if (fl)                                           cm = OBJ_W;     // scatter wins
    else if (b == NBATCH - 1 && curconf[hw] > SIL_TH) cm = 0.0f;      // silenced
    else                                              cm = NOOBJ_W;

    float local = 0.0f;
    if (epochp[0] > PRETRAIN) {
        float conf  = sigm(ob[18 * HWSZ + hw]);
        float tconf = sigm(db[18 * HWSZ + hw]);
        float dd = __builtin_amdgcn_sqrtf(cm) * (conf - tconf); // mimic sqrt-then-square
        local = 0.5f * dd * dd;
    }
    if (fl) {
#pragma unroll
        for (int k = 0; k < KP; ++k) {
            float xv = ob[(2 * k) * HWSZ + hw];
            float yv = ob[(2 * k + 1) * HWSZ + hw];
            float tx, ty;
            if (k == 0) {
                xv = sigm(xv); yv = sigm(yv);
                tx = sigm(db[hw]); ty = sigm(db[HWSZ + hw]);
            } else {
                tx = db[(k + 1) * HWSZ + hw];       // tx[k] = dt ch(k+1)
                ty = db[(k + 2) * HWSZ + hw];       // ty[k] = dt ch(k+2)
            }
            float dx = xv - tx, dy = yv - ty;
            local += 0.5f * (dx * dx + dy * dy);
        }
    }

    sdata[threadIdx.x] = local;
    __syncthreads();
    for (int s = 128; s > 0; s >>= 1) {
        if (threadIdx.x < s) sdata[threadIdx.x] += sdata[threadIdx.x + s];
        __syncthreads();
    }
    if (threadIdx.x == 0) partials[blockIdx.x] = sdata[0];
}

// ---------------------------------------------------------------------------
// Kernel R: deterministic final reduction of 4096 partials -> d_out[0]
// ---------------------------------------------------------------------------
__global__ void reduce_kernel(const float* __restrict__ partials,
                              float* __restrict__ outv, int nPart) {
    __shared__ float sdata[256];
    float s = 0.0f;
    for (int i = threadIdx.x; i < nPart; i += 256) s += partials[i];
    sdata[threadIdx.x] = s;
    __syncthreads();
    for (int m = 128; m > 0; m >>= 1) {
        if (threadIdx.x < m) sdata[threadIdx.x] += sdata[threadIdx.x + m];
        __syncthreads();
    }
    if (threadIdx.x == 0) outv[0] = sdata[0];
}

// ---------------------------------------------------------------------------
extern "C" void kernel_launch(void* const* d_in, const int* in_sizes, int n_in,
                              void* d_out, int out_size, void* d_ws, size_t ws_size,
                              hipStream_t stream) {
    const float* output = (const float*)d_in[0];
    const float* target = (const float*)d_in[1];
    const float* dtgt   = (const float*)d_in[2];
    const int*   epoch  = (const int*)d_in[3];
    float* outp = (float*)d_out;

    // workspace layout (all offsets 16B aligned)
    char* ws = (char*)d_ws;
    unsigned char* flagArr = (unsigned char*)ws;                    //  1,048,576 B
    float*  curconf  = (float*) (ws + 1048576);                     //     65,536 B
    float2* gt       = (float2*)(ws + 1048576 + 65536);             //      4,608 B
    float*  validf   = (float*) (ws + 1048576 + 65536 + 4608);      //        256 B
    float*  partials = (float*) (ws + 1048576 + 65536 + 4608 + 256);//     16,384 B

    zero_kernel<<<1024, 256, 0, stream>>>((unsigned int*)flagArr, NBATCH * HWSZ / 4);
    target_kernel<<<1, 64, 0, stream>>>(target, flagArr, gt, validf);
    curconf_kernel<<<128, 256, 0, stream>>>(output, gt, validf, curconf);
    loss_kernel<<<4096, 256, 0, stream>>>(output, dtgt, flagArr, curconf, epoch, partials);
    reduce_kernel<<<1, 256, 0, stream>>>(partials, outp, 4096);
}